// SparseGCNLayer_60069412601925
// MI455X (gfx1250) — compile-verified
//
#include <hip/hip_runtime.h>

typedef __attribute__((ext_vector_type(2))) float v2f;
typedef __attribute__((ext_vector_type(8))) float v8f;

#define N_IN  64
#define N_OUT 64

// ---------------------------------------------------------------------------
// Edge scatter: out[dst] += val * h[src].  16 threads per edge, float4 per
// thread. Native non-returning fp32 atomic adds at DEV scope -> executed at
// the L2 atomic units. h (25.6 MB) and out (25.6 MB) are both L2-resident
// (192 MB L2), so this phase runs at L2 rates, not HBM.
// Placed first in the file so the disasm snippet shows the atomic lowering.
// ---------------------------------------------------------------------------
__device__ __forceinline__ void atomic_fadd_dev(float* p, float v, int byte_off)
{
    switch (byte_off) {
    case 0:
        asm volatile("global_atomic_add_f32 %0, %1, off scope:SCOPE_DEV"
                     :: "v"(p), "v"(v) : "memory");
        break;
    case 4:
        asm volatile("global_atomic_add_f32 %0, %1, off offset:4 scope:SCOPE_DEV"
                     :: "v"(p), "v"(v) : "memory");
        break;
    case 8:
        asm volatile("global_atomic_add_f32 %0, %1, off offset:8 scope:SCOPE_DEV"
                     :: "v"(p), "v"(v) : "memory");
        break;
    default:
        asm volatile("global_atomic_add_f32 %0, %1, off offset:12 scope:SCOPE_DEV"
                     :: "v"(p), "v"(v) : "memory");
        break;
    }
}

__global__ __launch_bounds__(256) void gcn_scatter(
    const float* __restrict__ H, const float* __restrict__ eval,
    const int* __restrict__ esrc, const int* __restrict__ edst,
    float* __restrict__ out, int n_edges)
{
    const long long idx = (long long)blockIdx.x * 256 + threadIdx.x;
    const int e = (int)(idx >> 4);
    if (e >= n_edges) return;
    const int c = ((int)idx & 15) << 2;   // feature chunk [c, c+3]

    const float v  = eval[e];
    const float4 h = *(const float4*)&H[(size_t)esrc[e] * N_OUT + c];
    float* o = &out[(size_t)edst[e] * N_OUT + c];

    atomic_fadd_dev(o, v * h.x, 0);
    atomic_fadd_dev(o, v * h.y, 4);
    atomic_fadd_dev(o, v * h.z, 8);
    atomic_fadd_dev(o, v * h.w, 12);
}

// ---------------------------------------------------------------------------
// H = X @ W using V_WMMA_F32_16X16X4_F32.
// Block = 256 threads = 8 waves; each wave owns one 16-row tile of X and
// produces the full 16x64 output strip (4 accumulators of 16x16).
// ---------------------------------------------------------------------------
__global__ __launch_bounds__(256) void gcn_gemm_xw(
    const float* __restrict__ X, const float* __restrict__ W,
    float* __restrict__ H, int n_tiles)
{
    __shared__ float sW[N_IN * N_OUT];       // 16 KB
    __shared__ float sX[8][16 * N_IN];       // 8 waves * 4 KB = 32 KB

    const int tid  = threadIdx.x;
    const int wave = tid >> 5;               // wave32
    const int lane = tid & 31;
    const int tile = blockIdx.x * 8 + wave;
    const bool active = (tile < n_tiles);    // wave-uniform -> EXEC stays full

    // Cooperative load of W (4096 floats) as float4.
    for (int i = tid * 4; i < N_IN * N_OUT; i += 256 * 4)
        *(float4*)&sW[i] = *(const float4*)&W[i];

    // Per-wave load of its 16x64 X tile: rows are contiguous, so the tile is
    // 1024 contiguous floats -> fully coalesced float4 loads.
    if (active) {
        const float* xt = X + (size_t)tile * (16 * N_IN);
        float* st = &sX[wave][0];
        #pragma unroll
        for (int i = 0; i < 8; ++i) {
            const int off = i * 128 + lane * 4;
            *(float4*)&st[off] = *(const float4*)&xt[off];
        }
    }
    __syncthreads();

    if (active) {
        const int half = lane >> 4;   // 0: lanes 0-15, 1: lanes 16-31
        const int lo   = lane & 15;

        v8f acc[4] = {};              // 4 N-tiles of 16x16 f32
        const float* xs = &sX[wave][0];

        #pragma unroll
        for (int ks = 0; ks < 16; ++ks) {       // K = 64 in steps of 4
            // A fragment (16x4): lane half selects K pair {0,1} vs {2,3}
            const int kA = ks * 4 + half * 2;
            v2f a;
            a.x = xs[lo * N_IN + kA + 0];
            a.y = xs[lo * N_IN + kA + 1];
            #pragma unroll
            for (int nt = 0; nt < 4; ++nt) {
                // B fragment (4x16): VGPR v holds row K = v + 2*half
                v2f b;
                b.x = sW[(ks * 4 + half * 2 + 0) * N_OUT + nt * 16 + lo];
                b.y = sW[(ks * 4 + half * 2 + 1) * N_OUT + nt * 16 + lo];
                acc[nt] = __builtin_amdgcn_wmma_f32_16x16x4_f32(
                    false, a, false, b, (short)0, acc[nt], false, false);
            }
        }

        // D layout: VGPR v holds row (v + 8*half), column lo (per N-tile).
        float* ht = H + (size_t)tile * (16 * N_OUT);
        #pragma unroll
        for (int nt = 0; nt < 4; ++nt)
            #pragma unroll
            for (int v = 0; v < 8; ++v)
                ht[(v + 8 * half) * N_OUT + nt * 16 + lo] = acc[nt][v];
    }
}

// ---------------------------------------------------------------------------
// Zero the accumulation buffer (float4 stores).
// ---------------------------------------------------------------------------
__global__ __launch_bounds__(256) void gcn_zero(float4* __restrict__ out, int n4)
{
    const int i = blockIdx.x * 256 + threadIdx.x;
    if (i < n4) out[i] = make_float4(0.f, 0.f, 0.f, 0.f);
}

// ---------------------------------------------------------------------------
// In-place ReLU over the aggregated output.
// ---------------------------------------------------------------------------
__global__ __launch_bounds__(256) void gcn_relu(float4* __restrict__ out, int n4)
{
    const int i = blockIdx.x * 256 + threadIdx.x;
    if (i < n4) {
        float4 x = out[i];
        x.x = fmaxf(x.x, 0.f);
        x.y = fmaxf(x.y, 0.f);
        x.z = fmaxf(x.z, 0.f);
        x.w = fmaxf(x.w, 0.f);
        out[i] = x;
    }
}

// ---------------------------------------------------------------------------
extern "C" void kernel_launch(void* const* d_in, const int* in_sizes, int n_in,
                              void* d_out, int out_size, void* d_ws, size_t ws_size,
                              hipStream_t stream)
{
    const float* X    = (const float*)d_in[0];
    const float* W    = (const float*)d_in[1];
    const float* eval = (const float*)d_in[2];
    const int*   esrc = (const int*)d_in[3];
    const int*   edst = (const int*)d_in[4];
    float* out = (float*)d_out;
    float* H   = (float*)d_ws;               // n_nodes * 64 fp32 = 25.6 MB

    const int n_nodes = in_sizes[0] / N_IN;
    const int n_edges = in_sizes[2];
    const int n_tiles = (n_nodes + 15) / 16; // 6250 (exact for 100000 nodes)
    const int n4      = out_size / 4;

    gcn_zero<<<(n4 + 255) / 256, 256, 0, stream>>>((float4*)out, n4);

    gcn_gemm_xw<<<(n_tiles + 7) / 8, 256, 0, stream>>>(X, W, H, n_tiles);

    const long long work = (long long)n_edges * 16;
    gcn_scatter<<<(int)((work + 255) / 256), 256, 0, stream>>>(
        H, eval, esrc, edst, out, n_edges);

    gcn_relu<<<(n4 + 255) / 256, 256, 0, stream>>>((float4*)out, n4);
}